// LightIncXML_47296179863794
// MI455X (gfx1250) — compile-verified
//
#include <hip/hip_runtime.h>
#include <hip/hip_bf16.h>
#include <math.h>
#include <stdint.h>

typedef __attribute__((ext_vector_type(2))) float v2f;
typedef __attribute__((ext_vector_type(8))) float v8f;

#define B_DIM 32
#define HID   768
#define OUT5  2304
#define HDIM  512
#define NC    65536
#define MAXG  11
#define NL    670091
#define TOPK  10
#define CAND  110   // TOPK * MAXG

// Async global->LDS staging (CDNA5 GLOBAL_LOAD_ASYNC_TO_LDS_*, ASYNCcnt).
// Gated so a one-line revert restores the load+ds_store fallback.
#define USE_ASYNC_LDS 1

__device__ __forceinline__ float gelu_exact(float x) {
    // torch nn.GELU (exact): 0.5*x*(1+erf(x/sqrt(2)))
    return 0.5f * x * (1.0f + erff(x * 0.70710678118654752f));
}

// -------------------------------------------------------------------------
// D[32 x N] = GELU(A[32 x K] @ W[K x N] + bias[N])
// One wave per 16-column tile; each wave owns two 16x16 C tiles (rows 0-15
// and 16-31) so every B fragment feeds two v_wmma_f32_16x16x4_f32 ops.
// -------------------------------------------------------------------------
__global__ void wmma_gemm_gelu(const float* __restrict__ A,
                               const float* __restrict__ W,
                               const float* __restrict__ bias,
                               float* __restrict__ D,
                               int K, int N) {
    const int wave = (blockIdx.x * blockDim.x + threadIdx.x) >> 5;
    const int lane = threadIdx.x & 31;
    const int n0   = wave * 16;
    const int m    = lane & 15;          // row (A) / column (B,D) within tile
    const int kb   = (lane >> 4) * 2;    // K sub-offset held by this lane
    const int n    = n0 + m;

    v8f c0 = {}; v8f c1 = {};
    for (int k = 0; k < K; k += 4) {
        v2f a0 = *(const v2f*)&A[(size_t)m * K + k + kb];
        v2f a1 = *(const v2f*)&A[(size_t)(m + 16) * K + k + kb];
        v2f b;
        b.x = W[(size_t)(k + kb) * N + n];
        b.y = W[(size_t)(k + kb + 1) * N + n];
        c0 = __builtin_amdgcn_wmma_f32_16x16x4_f32(false, a0, false, b, (short)0, c0, false, false);
        c1 = __builtin_amdgcn_wmma_f32_16x16x4_f32(false, a1, false, b, (short)0, c1, false, false);
    }
    const float bv = bias[n];
    const int   mr = 8 * (lane >> 4);
#pragma unroll
    for (int r = 0; r < 8; ++r) {
        D[(size_t)(mr + r) * N + n]      = gelu_exact(c0[r] + bv);
        D[(size_t)(16 + mr + r) * N + n] = gelu_exact(c1[r] + bv);
    }
}

// -------------------------------------------------------------------------
// meta_logits[32 x 65536] = h[32 x 512] @ W_meta2[512 x 65536] + b_meta2
// HBM-bound: 134 MB weight stream. h is staged through LDS in 32x64 K-tiles
// (stride-66 padding -> conflict-free ds_load_b64 A fragments). The staging
// uses async global->LDS copies (ASYNCcnt) to skip the VGPR round-trip.
// 512 blocks x 8 waves; each wave owns one 16-column tile.
// -------------------------------------------------------------------------
#define KTILE 64
#define LDSST 66

__global__ void wmma_meta_gemm(const float* __restrict__ Hh,
                               const float* __restrict__ W,
                               const float* __restrict__ bias,
                               float* __restrict__ D) {
    __shared__ float lh[B_DIM * LDSST];

    const int wave = threadIdx.x >> 5;
    const int lane = threadIdx.x & 31;
    const int n0   = (blockIdx.x * 8 + wave) * 16;
    const int m    = lane & 15;
    const int kb   = (lane >> 4) * 2;
    const int n    = n0 + m;

    v8f c0 = {}; v8f c1 = {};
    for (int k0 = 0; k0 < HDIM; k0 += KTILE) {
        // cooperative stage of h[:, k0:k0+64] into LDS (8 floats/thread)
        {
            const int row = threadIdx.x >> 3;
            const int ks  = (threadIdx.x & 7) * 8;
#if USE_ASYNC_LDS
            // LDS byte offset = ptrtoint of the addrspace(3) pointer
            uint32_t lds_off =
                (uint32_t)(uintptr_t)(__attribute__((address_space(3))) float*)
                    &lh[row * LDSST + ks];
            uint64_t gaddr = (uint64_t)(uintptr_t)&Hh[(size_t)row * HDIM + k0 + ks];
#pragma unroll
            for (int j = 0; j < 4; ++j) {
                asm volatile("global_load_async_to_lds_b64 %0, %1, off"
                             :: "v"(lds_off + 8u * j), "v"(gaddr + 8u * j)
                             : "memory");
            }
            asm volatile("s_wait_asynccnt 0x0" ::: "memory");
#else
#pragma unroll
            for (int j = 0; j < 8; ++j)
                lh[row * LDSST + ks + j] = Hh[(size_t)row * HDIM + k0 + ks + j];
#endif
        }
        __syncthreads();

        if (k0 + KTILE < HDIM)  // pull next weight tile toward L2 early
            __builtin_prefetch(&W[(size_t)(k0 + KTILE + kb) * NC + n], 0, 1);

#pragma unroll 4
        for (int kk = 0; kk < KTILE; kk += 4) {
            v2f a0 = *(const v2f*)&lh[m * LDSST + kk + kb];
            v2f a1 = *(const v2f*)&lh[(m + 16) * LDSST + kk + kb];
            v2f b;
            const float* wp = &W[(size_t)(k0 + kk + kb) * NC + n];
            b.x = wp[0];
            b.y = wp[NC];
            c0 = __builtin_amdgcn_wmma_f32_16x16x4_f32(false, a0, false, b, (short)0, c0, false, false);
            c1 = __builtin_amdgcn_wmma_f32_16x16x4_f32(false, a1, false, b, (short)0, c1, false, false);
        }
        __syncthreads();
    }
    const float bv = bias[n];
    const int   mr = 8 * (lane >> 4);
#pragma unroll
    for (int r = 0; r < 8; ++r) {
        D[(size_t)(mr + r) * NC + n]      = c0[r] + bv;
        D[(size_t)(16 + mr + r) * NC + n] = c1[r] + bv;
    }
}

// -------------------------------------------------------------------------
// Top-10 per row of meta_logits (sigmoid is monotone -> topk on logits,
// sigmoid applied to winners). One block per row; non-destructive scan that
// skips already-chosen indices, so graph replays are deterministic.
// -------------------------------------------------------------------------
__global__ void topk_kernel(const float* __restrict__ logits,
                            float* __restrict__ tscore,
                            int* __restrict__ tidx) {
    __shared__ float sval[256];
    __shared__ int   sidx[256];
    __shared__ int   chosen[TOPK];

    const int row = blockIdx.x;
    const float* rl = logits + (size_t)row * NC;

    for (int t = 0; t < TOPK; ++t) {
        float bv = -INFINITY; int bi = NC;
        for (int j = threadIdx.x; j < NC; j += 256) {
            bool skip = false;
            for (int u = 0; u < t; ++u) skip |= (chosen[u] == j);
            if (skip) continue;
            float v = rl[j];
            if (v > bv || (v == bv && j < bi)) { bv = v; bi = j; }
        }
        sval[threadIdx.x] = bv; sidx[threadIdx.x] = bi;
        __syncthreads();
        for (int s = 128; s > 0; s >>= 1) {
            if (threadIdx.x < s) {
                float ov = sval[threadIdx.x + s]; int oi = sidx[threadIdx.x + s];
                if (ov > sval[threadIdx.x] ||
                    (ov == sval[threadIdx.x] && oi < sidx[threadIdx.x])) {
                    sval[threadIdx.x] = ov; sidx[threadIdx.x] = oi;
                }
            }
            __syncthreads();
        }
        if (threadIdx.x == 0) {
            chosen[t] = sidx[0];
            tidx[row * TOPK + t]   = sidx[0];
            tscore[row * TOPK + t] = 1.0f / (1.0f + expf(-sval[0]));
        }
        __syncthreads();
    }
}

// -------------------------------------------------------------------------
// Final stage: gather group_y / embed, batched dot against h_ext, sigmoid
// with the logit==0 -> -inf padding rule, combine with candidate scores.
// One wave (32 lanes) per (batch, candidate) dot product of length 512.
// -------------------------------------------------------------------------
__global__ void score_kernel(const float* __restrict__ hext,
                             const float* __restrict__ embed,
                             const int*   __restrict__ group_y,
                             const float* __restrict__ tscore,
                             const int*   __restrict__ tidx,
                             float* __restrict__ out) {
    const int b    = blockIdx.x;
    const int wave = threadIdx.x >> 5;
    const int lane = threadIdx.x & 31;
    const float* hb = hext + (size_t)b * HDIM;

    for (int c = wave; c < CAND; c += 8) {
        const int   g    = tidx[b * TOPK + c / MAXG];
        const float gs   = tscore[b * TOPK + c / MAXG];
        const int   cand = group_y[(size_t)g * MAXG + (c % MAXG)];
        const float* er  = embed + (size_t)cand * HDIM;

        float s = 0.0f;
        for (int h = lane; h < HDIM; h += 32)
            s += er[h] * hb[h];
#pragma unroll
        for (int off = 16; off > 0; off >>= 1)
            s += __shfl_xor(s, off, 32);

        if (lane == 0) {
            // padding embed row is all-zero -> dot == 0 -> sigmoid(-inf) == 0
            const float cs   = (s == 0.0f) ? 0.0f : 1.0f / (1.0f + expf(-s));
            const float comb = (cand != NL) ? cs * gs : 0.0f;
            out[(size_t)b * CAND + c]                      = (float)cand;  // cands
            out[(size_t)B_DIM * CAND + b * CAND + c]       = cs;           // cscores
            out[2 * (size_t)B_DIM * CAND + b * CAND + c]   = comb;         // comb
        }
    }
}

extern "C" void kernel_launch(void* const* d_in, const int* in_sizes, int n_in,
                              void* d_out, int out_size, void* d_ws, size_t ws_size,
                              hipStream_t stream) {
    const float* hid_first = (const float*)d_in[0];
    const float* hid_last5 = (const float*)d_in[1];
    const float* W_meta1   = (const float*)d_in[2];
    const float* b_meta1   = (const float*)d_in[3];
    const float* W_meta2   = (const float*)d_in[4];
    const float* b_meta2   = (const float*)d_in[5];
    const float* W_ext     = (const float*)d_in[6];
    const float* b_ext     = (const float*)d_in[7];
    const float* embed     = (const float*)d_in[8];
    const int*   group_y   = (const int*)d_in[9];   // harness passes integers as int32
    float* out = (float*)d_out;

    // workspace carve-up
    char*  ws     = (char*)d_ws;
    float* h      = (float*)ws;                                   // 32*512*4   = 64 KB
    float* hext   = (float*)(ws + (64 << 10));                    // 64 KB
    float* mlog   = (float*)(ws + (128 << 10));                   // 32*65536*4 = 8 MB
    float* tscore = (float*)(ws + (128 << 10) + (8u << 20));      // 1.25 KB
    int*   tidx   = (int*)  (ws + (128 << 10) + (8u << 20) + 2048);

    // 1) h = GELU(hid_first @ W_meta1 + b_meta1)      [32 x 512]
    wmma_gemm_gelu<<<(HDIM / 16) / 8, 256, 0, stream>>>(hid_first, W_meta1, b_meta1, h, HID, HDIM);
    // 2) h_ext = GELU(hid_last5 @ W_ext + b_ext)      [32 x 512] (independent)
    wmma_gemm_gelu<<<(HDIM / 16) / 8, 256, 0, stream>>>(hid_last5, W_ext, b_ext, hext, OUT5, HDIM);
    // 3) meta_logits = h @ W_meta2 + b_meta2          [32 x 65536] (HBM-bound)
    wmma_meta_gemm<<<(NC / 16) / 8, 256, 0, stream>>>(h, W_meta2, b_meta2, mlog);
    // 4) per-row top-10 (sigmoid of winners)
    topk_kernel<<<B_DIM, 256, 0, stream>>>(mlog, tscore, tidx);
    // 5) gather + batched matvec + sigmoid/combine -> (cands, cscores, comb)
    score_kernel<<<B_DIM, 256, 0, stream>>>(hext, embed, group_y, tscore, tidx, out);
}